// BahdanauAttention_90615220011858
// MI455X (gfx1250) — compile-verified
//
#include <hip/hip_runtime.h>
#include <hip/hip_bf16.h>

// Bahdanau attention, B=4, Te=Td=256, De=Dd=512, fp32.
#define B_  4
#define TE_ 256
#define TD_ 256
#define DE_ 512
#define DD_ 512

typedef __attribute__((ext_vector_type(2))) float v2f;
typedef __attribute__((ext_vector_type(4))) float v4f;
typedef __attribute__((ext_vector_type(8))) float v8f;

#if __has_builtin(__builtin_amdgcn_tanhf)
__device__ __forceinline__ float fast_tanh(float x) { return __builtin_amdgcn_tanhf(x); }
#elif __has_builtin(__builtin_amdgcn_tanh_f32)
__device__ __forceinline__ float fast_tanh(float x) { return __builtin_amdgcn_tanh_f32(x); }
#else
__device__ __forceinline__ float fast_tanh(float x) { return tanhf(x); }
#endif

#if __has_builtin(__builtin_amdgcn_exp2f)
__device__ __forceinline__ float fast_exp2(float x) { return __builtin_amdgcn_exp2f(x); }
#else
__device__ __forceinline__ float fast_exp2(float x) { return exp2f(x); }
#endif

// ---------------------------------------------------------------------------
// C[b] (MxN) = A[b] (MxK) @ B[b] (KxN), all row-major fp32.
// One wave32 per 16x16 output tile; K-loop in steps of 4 using
// V_WMMA_F32_16X16X4_F32 (full-precision fp32 matrix pipe on CDNA5).
// Fragment layouts per CDNA5 ISA:
//   A 16x4 : lane r (=M row), half=lane>>4 : a[v] = A[r][2*half + v]
//   B 4x16 : lane r (=N col)               : b[v] = B[2*half + v][r]
//   C 16x16: acc[i] = C[i + 8*half][r]
// ---------------------------------------------------------------------------
__global__ __launch_bounds__(32) void wmma_gemm_f32(
    const float* __restrict__ A, const float* __restrict__ Bm,
    float* __restrict__ C, int K, int N,
    long sA, long sB, long sC)
{
    const int lane = threadIdx.x & 31;
    const int half = lane >> 4;
    const int r    = lane & 15;

    const float* Ab = A  + (long)blockIdx.z * sA;
    const float* Bb = Bm + (long)blockIdx.z * sB;
    float*       Cb = C  + (long)blockIdx.z * sC;

    const int m0 = blockIdx.x * 16;
    const int n0 = blockIdx.y * 16;

    const float* aRow = Ab + (long)(m0 + r) * K;   // this lane's A row
    const float* bCol = Bb + n0 + r;               // this lane's B column

    v8f acc = {};
    for (int k0 = 0; k0 < K; k0 += 4) {
        // contiguous pair -> single global_load_b64
        v2f a = *(const v2f*)(aRow + k0 + 2 * half);
        v2f b;
        b.x = bCol[(long)(k0 + 2 * half + 0) * N];
        b.y = bCol[(long)(k0 + 2 * half + 1) * N];
        acc = __builtin_amdgcn_wmma_f32_16x16x4_f32(
            /*neg_a=*/false, a, /*neg_b=*/false, b,
            /*c_mod=*/(short)0, acc, /*reuse_a=*/false, /*reuse_b=*/false);
    }

#pragma unroll
    for (int i = 0; i < 8; ++i)
        Cb[(long)(m0 + i + 8 * half) * N + n0 + r] = acc[i];
}

// ---------------------------------------------------------------------------
// Fused energy + softmax per (b, t):
//   e[te] = sum_d Va[d] * tanh(Uh[b,te,d] + Ws[b,t,d])   (v_tanh_f32)
//   attn[b,t,:] = softmax(e)
// Never materializes the 537MB broadcast tensor; Ws row + Va live in LDS,
// Uh rows stream through the 192MB L2 (entire problem is L2-resident).
// 256 threads = 8 waves; each wave owns te = wave, wave+8, ...
// All data movement is float4 (global_load_b128 / ds_load_b128): 3 memory
// ops per 4 tanh instead of 12, so the TRANS pipe (v_tanh_f32) is the limit.
// ---------------------------------------------------------------------------
__global__ __launch_bounds__(256) void energy_softmax_kernel(
    const float* __restrict__ Uh,   // [B,TE,DE]
    const float* __restrict__ Ws,   // [B,TD,DE]
    const float* __restrict__ Va,   // [DE]
    float* __restrict__ attn)       // [B,TD,TE]
{
    __shared__ v4f ws_s[DE_ / 4];   // 512 floats
    __shared__ v4f va_s[DE_ / 4];   // 512 floats
    __shared__ float e_s[TE_];
    __shared__ float red[9];

    const int t   = blockIdx.x;
    const int b   = blockIdx.y;
    const int tid = threadIdx.x;

    if (tid < DE_ / 4) {
        ws_s[tid] = ((const v4f*)(Ws + ((long)b * TD_ + t) * DE_))[tid];
        va_s[tid] = ((const v4f*)Va)[tid];
    }
    __syncthreads();

    const int lane = tid & 31;
    const int wv   = tid >> 5;

    for (int te = wv; te < TE_; te += 8) {
        const v4f* uh4 = (const v4f*)(Uh + ((long)b * TE_ + te) * DE_);
        float p = 0.f;
#pragma unroll
        for (int i = 0; i < DE_ / 128; ++i) {      // 4 iterations
            const int d4 = lane + 32 * i;          // float4 index
            const v4f u = uh4[d4];                 // global_load_b128
            const v4f w = ws_s[d4];                // ds_load_b128
            const v4f a = va_s[d4];                // ds_load_b128
            p += a.x * fast_tanh(u.x + w.x);
            p += a.y * fast_tanh(u.y + w.y);
            p += a.z * fast_tanh(u.z + w.z);
            p += a.w * fast_tanh(u.w + w.w);
        }
        for (int off = 16; off; off >>= 1)
            p += __shfl_down(p, off, 32);
        if (lane == 0) e_s[te] = p;
    }
    __syncthreads();

    // --- softmax over e_s (one element per thread, TE_ == blockDim) ---
    const float v = e_s[tid];

    float m = v;
    for (int off = 16; off; off >>= 1) m = fmaxf(m, __shfl_down(m, off, 32));
    if (lane == 0) red[wv] = m;
    __syncthreads();
    if (tid == 0) {
        float mm = red[0];
        for (int i = 1; i < 8; ++i) mm = fmaxf(mm, red[i]);
        red[8] = mm;
    }
    __syncthreads();
    m = red[8];

    const float p = fast_exp2((v - m) * 1.4426950408889634f);
    float s = p;
    for (int off = 16; off; off >>= 1) s += __shfl_down(s, off, 32);
    if (lane == 0) red[wv] = s;
    __syncthreads();
    if (tid == 0) {
        float ss = red[0];
        for (int i = 1; i < 8; ++i) ss += red[i];
        red[8] = ss;
    }
    __syncthreads();

    attn[((long)b * TD_ + t) * TE_ + tid] = p / red[8];
}

// ---------------------------------------------------------------------------
// Launch sequence:
//   1) Uh = enc @ U_a           (WMMA f32)
//   2) Ws = dec @ W_a           (WMMA f32)
//   3) fused energy+softmax  -> attn (second half of d_out)
//   4) context = attn @ enc     (WMMA f32) -> first half of d_out
// Workspace: Uh (2MB) + Ws (2MB) in d_ws.
// ---------------------------------------------------------------------------
extern "C" void kernel_launch(void* const* d_in, const int* in_sizes, int n_in,
                              void* d_out, int out_size, void* d_ws, size_t ws_size,
                              hipStream_t stream) {
    const float* enc = (const float*)d_in[0];   // [B,TE,DE]
    const float* dec = (const float*)d_in[1];   // [B,TD,DD]
    const float* Ua  = (const float*)d_in[2];   // [DE,DE]
    const float* Wa  = (const float*)d_in[3];   // [DD,DE]
    const float* Va  = (const float*)d_in[4];   // [DE]

    float* out     = (float*)d_out;
    float* context = out;                                   // [B,TD,DE]
    float* attn    = out + (long)B_ * TD_ * DE_;            // [B,TD,TE]

    float* Uh  = (float*)d_ws;                              // [B,TE,DE]
    float* Wsb = Uh + (long)B_ * TE_ * DE_;                 // [B,TD,DE]

    const dim3 wblk(32);

    // Uh = enc @ Ua : M=TE, K=DE, N=DE (Ua shared across batch -> stride 0)
    wmma_gemm_f32<<<dim3(TE_ / 16, DE_ / 16, B_), wblk, 0, stream>>>(
        enc, Ua, Uh, DE_, DE_, (long)TE_ * DE_, 0L, (long)TE_ * DE_);

    // Ws = dec @ Wa : M=TD, K=DD, N=DE
    wmma_gemm_f32<<<dim3(TD_ / 16, DE_ / 16, B_), wblk, 0, stream>>>(
        dec, Wa, Wsb, DD_, DE_, (long)TD_ * DD_, 0L, (long)TD_ * DE_);

    // energies + softmax -> attn
    energy_softmax_kernel<<<dim3(TD_, B_), dim3(256), 0, stream>>>(
        Uh, Wsb, Va, attn);

    // context = attn @ enc : M=TD, K=TE, N=DE
    wmma_gemm_f32<<<dim3(TD_ / 16, DE_ / 16, B_), wblk, 0, stream>>>(
        attn, enc, context, TE_, DE_, (long)TD_ * TE_, (long)TE_ * DE_, (long)TD_ * DE_);
}